// ContrastiveGCN_45827301048543
// MI455X (gfx1250) — compile-verified
//
#include <hip/hip_runtime.h>
#include <hip/hip_bf16.h>

typedef __attribute__((ext_vector_type(2))) float v2f;
typedef __attribute__((ext_vector_type(8))) float v8f;

#define N_NODES 50000
#define N_EDGES 800000

// ---------------- degree / normalization ----------------

__global__ void deg_init_kernel(float* __restrict__ deg, int n) {
    int i = blockIdx.x * blockDim.x + threadIdx.x;
    if (i < n) deg[i] = 1.0f;           // self-loop contributes 1
}

__global__ void deg_count_kernel(const long long* __restrict__ dst,
                                 float* __restrict__ deg, int ne) {
    int e = blockIdx.x * blockDim.x + threadIdx.x;
    if (e < ne) atomicAdd(&deg[(int)dst[e]], 1.0f);
}

__global__ void dinv_kernel(float* __restrict__ deg, int n) {
    int i = blockIdx.x * blockDim.x + threadIdx.x;
    if (i < n) deg[i] = rsqrtf(deg[i]);  // in-place deg -> deg^-1/2
}

// ---------------- propagation: out = A_hat @ h ----------------
// Self-loop term: out[i] = h[i] * dinv[i]^2   (init, fully overwrites out)
__global__ void self_init_kernel(const float* __restrict__ h,
                                 const float* __restrict__ dinv,
                                 float* __restrict__ out,
                                 int n_nodes, int lgC) {
    size_t idx = (size_t)blockIdx.x * blockDim.x + threadIdx.x;   // float4 index
    size_t total4 = ((size_t)n_nodes << lgC) >> 2;
    if (idx >= total4) return;
    int node = (int)(idx >> (lgC - 2));
    float w = dinv[node]; w *= w;
    float4 v = ((const float4*)h)[idx];
    v.x *= w; v.y *= w; v.z *= w; v.w *= w;
    ((float4*)out)[idx] = v;
}

// Edge term: out[dst] += h[src] * dinv[src]*dinv[dst]  (f32 atomics, resolve in L2)
__global__ void edge_scatter_kernel(const long long* __restrict__ src,
                                    const long long* __restrict__ dst,
                                    const float* __restrict__ dinv,
                                    const float* __restrict__ h,
                                    float* __restrict__ out,
                                    int n_edges, int lgC4) {
    int t = blockIdx.x * blockDim.x + threadIdx.x;
    int e = t >> lgC4;
    if (e >= n_edges) return;
    int c4 = t & ((1 << lgC4) - 1);
    int s = (int)src[e];
    int d = (int)dst[e];
    float nrm = dinv[s] * dinv[d];
    int C = 4 << lgC4;
    float4 v = ((const float4*)(h + (size_t)s * C))[c4];
    float* op = out + (size_t)d * C + (c4 << 2);
    atomicAdd(op + 0, v.x * nrm);
    atomicAdd(op + 1, v.y * nrm);
    atomicAdd(op + 2, v.z * nrm);
    atomicAdd(op + 3, v.w * nrm);
}

// ---------------- dense GEMM via V_WMMA_F32_16X16X4_F32 ----------------
// out[M,N] = A1[M,K]@B1[K,N] (+ A2@B2) + bias, optional relu.
// One wave32 per 16x16 output tile; M,N multiples of 16, K multiple of 4.
// f32 A layout : lanes 0-15 hold K={0,1}, lanes 16-31 hold K={2,3}, M = lane&15
// f32 B layout : lanes 0-15 hold K rows {0,1}, lanes 16-31 rows {2,3}, N = lane&15
// f32 C layout : VGPR v -> row v (lanes 0-15) / row v+8 (lanes 16-31), col = lane&15
__global__ void __launch_bounds__(128)
gemm_wmma_f32_kernel(const float* __restrict__ A1, const float* __restrict__ B1,
                     const float* __restrict__ A2, const float* __restrict__ B2,
                     const float* __restrict__ bias, float* __restrict__ out,
                     int M, int K, int N, int do_relu) {
    const int lane = threadIdx.x & 31;
    const int tile = blockIdx.x * (blockDim.x >> 5) + (threadIdx.x >> 5);
    const int tilesN = N >> 4;
    const int tilesM = M >> 4;
    if (tile >= tilesM * tilesN) return;
    const int tm = tile / tilesN;
    const int tn = tile - tm * tilesN;
    const int col  = lane & 15;
    const int hi   = lane >> 4;          // 0 or 1
    const int koff = hi << 1;            // 0 or 2
    const int m = tm * 16 + col;
    const int n = tn * 16 + col;

    v8f acc = {};
    {
        const float* a = A1 + (size_t)m * K + koff;
        const float* b = B1 + (size_t)koff * N + n;
        for (int k = 0; k < K; k += 4) {
            v2f av; av.x = a[0]; av.y = a[1];
            v2f bv; bv.x = b[0]; bv.y = b[N];
            acc = __builtin_amdgcn_wmma_f32_16x16x4_f32(
                false, av, false, bv, (short)0, acc, false, false);
            a += 4; b += (size_t)4 * N;
        }
    }
    if (A2 != nullptr) {
        const float* a = A2 + (size_t)m * K + koff;
        const float* b = B2 + (size_t)koff * N + n;
        for (int k = 0; k < K; k += 4) {
            v2f av; av.x = a[0]; av.y = a[1];
            v2f bv; bv.x = b[0]; bv.y = b[N];
            acc = __builtin_amdgcn_wmma_f32_16x16x4_f32(
                false, av, false, bv, (short)0, acc, false, false);
            a += 4; b += (size_t)4 * N;
        }
    }
    const float bb = bias ? bias[n] : 0.0f;
    const int rbase = tm * 16 + (hi << 3);
#pragma unroll
    for (int v = 0; v < 8; ++v) {
        float r = acc[v] + bb;
        if (do_relu) r = fmaxf(r, 0.0f);
        out[(size_t)(rbase + v) * N + n] = r;
    }
}

// ---------------- tiny coordinate head: [n,16] @ [16,2] + b ----------------
__global__ void coord_head_kernel(const float* __restrict__ t2,
                                  const float* __restrict__ Wc2,
                                  const float* __restrict__ bc2,
                                  float* __restrict__ coords, int n) {
    int i = blockIdx.x * blockDim.x + threadIdx.x;
    if (i >= n) return;
    float a0 = bc2[0], a1 = bc2[1];
    const float* row = t2 + (size_t)i * 16;
#pragma unroll
    for (int k = 0; k < 16; ++k) {
        float v = row[k];
        a0 += v * Wc2[k * 2 + 0];
        a1 += v * Wc2[k * 2 + 1];
    }
    coords[(size_t)i * 2 + 0] = a0;
    coords[(size_t)i * 2 + 1] = a1;
}

// ---------------- launch ----------------

extern "C" void kernel_launch(void* const* d_in, const int* in_sizes, int n_in,
                              void* d_out, int out_size, void* d_ws, size_t ws_size,
                              hipStream_t stream) {
    const float*      x    = (const float*)d_in[0];
    const long long*  ei   = (const long long*)d_in[1];
    const long long*  esrc = ei;
    const long long*  edst = ei + N_EDGES;
    const float* W1_1 = (const float*)d_in[2];
    const float* W1_2 = (const float*)d_in[3];
    const float* b1   = (const float*)d_in[4];
    const float* W2_1 = (const float*)d_in[5];
    const float* W2_2 = (const float*)d_in[6];
    const float* b2   = (const float*)d_in[7];
    const float* Wp1  = (const float*)d_in[8];
    const float* bp1  = (const float*)d_in[9];
    const float* Wp2  = (const float*)d_in[10];
    const float* bp2  = (const float*)d_in[11];
    const float* Wc1  = (const float*)d_in[12];
    const float* bc1  = (const float*)d_in[13];
    const float* Wc2  = (const float*)d_in[14];
    const float* bc2  = (const float*)d_in[15];

    float* ws = (float*)d_ws;
    size_t o = 0;
    float* dinv = ws + o; o += 50048;                 // deg, then deg^-1/2 in place
    float* p1   = ws + o; o += (size_t)N_NODES * 128; // A_hat   x
    float* p2   = ws + o; o += (size_t)N_NODES * 128; // A_hat^2 x
    float* h1   = ws + o; o += (size_t)N_NODES * 64;
    float* h2   = ws + o; o += (size_t)N_NODES * 32;
    float* q1 = p1;   // A_hat   h1 (reuse)
    float* q2 = p2;   // A_hat^2 h1 (reuse)
    float* t1 = p1;   // relu(h2@Wp1+bp1) (reuse)
    float* t2 = p2;   // relu(h2@Wc1+bc1) (reuse)

    float* coords = (float*)d_out;                 // [50000,2]
    float* z      = (float*)d_out + 2 * N_NODES;   // [50000,32]

    const int T = 256;

    // normalization
    deg_init_kernel<<<(N_NODES + T - 1) / T, T, 0, stream>>>(dinv, N_NODES);
    deg_count_kernel<<<(N_EDGES + T - 1) / T, T, 0, stream>>>(edst, dinv, N_EDGES);
    dinv_kernel<<<(N_NODES + T - 1) / T, T, 0, stream>>>(dinv, N_NODES);

    // layer 1 propagations (C=128 -> lgC=7, lgC4=5)
    {
        size_t tot4 = (size_t)N_NODES * 128 / 4;
        self_init_kernel<<<(unsigned)((tot4 + T - 1) / T), T, 0, stream>>>(x, dinv, p1, N_NODES, 7);
        size_t et = (size_t)N_EDGES << 5;
        edge_scatter_kernel<<<(unsigned)((et + T - 1) / T), T, 0, stream>>>(esrc, edst, dinv, x, p1, N_EDGES, 5);
        self_init_kernel<<<(unsigned)((tot4 + T - 1) / T), T, 0, stream>>>(p1, dinv, p2, N_NODES, 7);
        edge_scatter_kernel<<<(unsigned)((et + T - 1) / T), T, 0, stream>>>(esrc, edst, dinv, p1, p2, N_EDGES, 5);
    }
    // h1 = relu(p1@W1_1 + p2@W1_2 + b1)   M=50000 K=128 N=64
    {
        int tiles = (N_NODES / 16) * (64 / 16);
        gemm_wmma_f32_kernel<<<(tiles + 3) / 4, 128, 0, stream>>>(
            p1, W1_1, p2, W1_2, b1, h1, N_NODES, 128, 64, 1);
    }
    // layer 2 propagations (C=64 -> lgC=6, lgC4=4)
    {
        size_t tot4 = (size_t)N_NODES * 64 / 4;
        self_init_kernel<<<(unsigned)((tot4 + T - 1) / T), T, 0, stream>>>(h1, dinv, q1, N_NODES, 6);
        size_t et = (size_t)N_EDGES << 4;
        edge_scatter_kernel<<<(unsigned)((et + T - 1) / T), T, 0, stream>>>(esrc, edst, dinv, h1, q1, N_EDGES, 4);
        self_init_kernel<<<(unsigned)((tot4 + T - 1) / T), T, 0, stream>>>(q1, dinv, q2, N_NODES, 6);
        edge_scatter_kernel<<<(unsigned)((et + T - 1) / T), T, 0, stream>>>(esrc, edst, dinv, q1, q2, N_EDGES, 4);
    }
    // h2 = relu(q1@W2_1 + q2@W2_2 + b2)   K=64 N=32
    {
        int tiles = (N_NODES / 16) * (32 / 16);
        gemm_wmma_f32_kernel<<<(tiles + 3) / 4, 128, 0, stream>>>(
            q1, W2_1, q2, W2_2, b2, h2, N_NODES, 64, 32, 1);
    }
    // projection head: z = relu(h2@Wp1+bp1)@Wp2 + bp2    (written straight to d_out)
    {
        int tiles = (N_NODES / 16) * (32 / 16);
        gemm_wmma_f32_kernel<<<(tiles + 3) / 4, 128, 0, stream>>>(
            h2, Wp1, nullptr, nullptr, bp1, t1, N_NODES, 32, 32, 1);
        gemm_wmma_f32_kernel<<<(tiles + 3) / 4, 128, 0, stream>>>(
            t1, Wp2, nullptr, nullptr, bp2, z, N_NODES, 32, 32, 0);
    }
    // coordinate head: t2 = relu(h2@Wc1+bc1) [K=32,N=16]; coords = t2@Wc2 + bc2
    {
        int tiles = (N_NODES / 16) * (16 / 16);
        gemm_wmma_f32_kernel<<<(tiles + 3) / 4, 128, 0, stream>>>(
            h2, Wc1, nullptr, nullptr, bc1, t2, N_NODES, 32, 16, 1);
        coord_head_kernel<<<(N_NODES + T - 1) / T, T, 0, stream>>>(t2, Wc2, bc2, coords, N_NODES);
    }
}